// ConditionedCTKoopmanTransition_30305289240725
// MI455X (gfx1250) — compile-verified
//
#include <hip/hip_runtime.h>
#include <hip/hip_bf16.h>

// ---------------------------------------------------------------------------
// ConditionedCTKoopmanTransition — CDNA5 (gfx1250) implementation.
//
// Phase 1: M = A_ct * dt / 2^8 built on device (skew unpack + softplus diag).
// Phase 2: E = e^M, P = phi1(M) by order-8 Taylor (Horner, GEMM form), then
//          8 doubling steps: P <- P(E+I)/2, E <- E^2.  ==> A_bar = E,
//          phi1(A_ct*dt) = P, B_bar = dt * P @ B_ct.   (No LU solve needed.)
//          Then one-time transposes: A_bar^T, B_bar^T, C^T (pad 64), dt*D^T.
// Phase 3: 256 sequential scan steps, each a 512x512x512 fp32 WMMA GEMM
//          (z@A_bar^T + u@B_bar^T) plus the observation GEMM (z@C^T + dt*u@D^T),
//          all plain row-major GEMMs against the pre-transposed operators.
//
// All matmuls use V_WMMA_F32_16X16X4_F32 (wave32 fp32 matrix pipe) to keep the
// 256-step recurrence numerically faithful to the fp32 reference.
// LDS layout: A panel row stride 34 (even -> aligned ds_load_b64 fragments),
// B panel k-pair interleaved so each B fragment is a single ds_load_b64.
// ---------------------------------------------------------------------------

#define D_DYN 512
#define U_DIM 32
#define N_OBS 50
#define NPAD 64          // padded observation columns
#define BATCH 512
#define T_STEPS 256
#define NSQ 8            // scaling/squaring count (2^8)

typedef __attribute__((ext_vector_type(2))) float v2f;
typedef __attribute__((ext_vector_type(8))) float v8f;

__device__ __forceinline__ v8f wmma4(v2f a, v2f b, v8f c) {
  // D(16x16,f32) = A(16x4,f32) x B(4x16,f32) + C
  return __builtin_amdgcn_wmma_f32_16x16x4_f32(
      /*neg_a=*/false, a, /*neg_b=*/false, b,
      /*c_mod=*/(short)0, c, /*reuse_a=*/false, /*reuse_b=*/false);
}

// Per-lane fragment indexing (ISA 7.12.2, 32-bit operands, wave32):
//   A 16x4 : vgpr0 = A[row=lane%16][k0], vgpr1 = A[row][k0+1], k0=(lane>=16)?2:0
//   B 4x16 : vgpr0 = B[k0][col=lane%16], vgpr1 = B[k0+1][col]
//   C 16x16: vgpr r = C[r + (lane>=16 ? 8:0)][lane%16]
#define ASTRIDE 34  // even row stride: 8B-aligned A fragments, distinct banks

// One k-step (K=4) of WMMA accumulation from LDS panels.
// As: [BM][ASTRIDE] floats. Bs: k-pair interleaved, pair (k2,col) at
// ((k2*BNP + col)*2).  RT row tiles x CTN col tiles per wave.
template <int RT, int CTN, int BNP>
__device__ __forceinline__ void mma_step(const float* __restrict__ As,
                                         const float* __restrict__ Bs,
                                         int lane, int kk, int wr, int wc,
                                         v8f (&acc)[RT][CTN]) {
  const int kr = kk + ((lane >> 4) << 1);  // even
  v2f a[RT], b[CTN];
#pragma unroll
  for (int rt = 0; rt < RT; ++rt) {
    int row = wr + rt * 16 + (lane & 15);
    a[rt] = *(const v2f*)&As[row * ASTRIDE + kr];
  }
#pragma unroll
  for (int ct = 0; ct < CTN; ++ct) {
    int col = wc + ct * 16 + (lane & 15);
    b[ct] = *(const v2f*)&Bs[((kr >> 1) * BNP + col) * 2];
  }
#pragma unroll
  for (int rt = 0; rt < RT; ++rt)
#pragma unroll
    for (int ct = 0; ct < CTN; ++ct)
      acc[rt][ct] = wmma4(a[rt], b[ct], acc[rt][ct]);
}

// ---------------------------------------------------------------------------
// Elementwise kernels
// ---------------------------------------------------------------------------

// M[i][j] = (dt/2^NSQ) * ( skew upper / -skew lower / -softplus(gamma) diag )
__global__ __launch_bounds__(256) void build_M_kernel(
    const float* __restrict__ skew, const float* __restrict__ gamma,
    const float* __restrict__ dt, float* __restrict__ Mb) {
  int idx = blockIdx.x * 256 + threadIdx.x;  // 512*512 total
  int i = idx >> 9, j = idx & 511;
  float v;
  if (i < j) {
    v = skew[(size_t)i * (2 * D_DYN - i - 1) / 2 + (j - i - 1)];
  } else if (i > j) {
    v = -skew[(size_t)j * (2 * D_DYN - j - 1) / 2 + (i - j - 1)];
  } else {
    float g = gamma[i];
    v = -(fmaxf(g, 0.0f) + log1pf(expf(-fabsf(g))));  // -softplus(g)
  }
  Mb[idx] = v * (dt[0] * (1.0f / 256.0f));
}

__global__ __launch_bounds__(256) void set_identity_kernel(
    float* __restrict__ buf, float c) {
  int idx = blockIdx.x * 256 + threadIdx.x;
  int i = idx >> 9, j = idx & 511;
  buf[idx] = (i == j) ? c : 0.0f;
}

// dst[src_cols][dst_cols]: dst[r][c] = (c < src_rows ? s*src[c][r] : 0)
// s = dtp ? dtp[0] : 1.   Grid covers src_cols*dst_cols elements.
__global__ __launch_bounds__(256) void transpose_pad_kernel(
    const float* __restrict__ src, float* __restrict__ dst,
    int src_rows, int src_cols, int dst_cols, const float* __restrict__ dtp) {
  int idx = blockIdx.x * 256 + threadIdx.x;
  int r = idx / dst_cols;  // source column
  int c = idx % dst_cols;  // source row (maybe padded)
  float s = dtp ? dtp[0] : 1.0f;
  dst[idx] = (c < src_rows) ? s * src[(size_t)c * src_cols + r] : 0.0f;
}

// ---------------------------------------------------------------------------
// Panel fill helpers (device inline, 256 threads)
// ---------------------------------------------------------------------------
__device__ __forceinline__ void fill_A_panel(float* __restrict__ As,
                                             const float* __restrict__ Ag,
                                             int bi, int k0, int lda, int tid) {
  for (int i = tid; i < 128 * 32; i += 256) {
    int r = i >> 5, c = i & 31;
    As[r * ASTRIDE + c] = Ag[(size_t)(bi + r) * lda + k0 + c];
  }
}

template <int BN, int BNP>
__device__ __forceinline__ void fill_B_panel(float* __restrict__ Bs,
                                             const float* __restrict__ Bg,
                                             int k0, int bj, int ldb, int tid) {
  for (int i = tid; i < 32 * BN; i += 256) {
    int r = i / BN, c = i % BN;
    Bs[((r >> 1) * BNP + c) * 2 + (r & 1)] =
        Bg[(size_t)(k0 + r) * ldb + bj + c];
  }
}

// ---------------------------------------------------------------------------
// Generic 512x512x512 fp32 WMMA GEMM: D = alpha*(A@B) + betaI*I + addScale*addMat
// grid = 16 WGs, 256 threads (8 waves); 128x128 tile; wave = 32x64 region.
// ---------------------------------------------------------------------------
__global__ __launch_bounds__(256) void gemm512_kernel(
    const float* __restrict__ A, const float* __restrict__ B,
    float* __restrict__ D, float alpha, float betaI,
    const float* __restrict__ addMat, float addScale) {
  const int d = D_DYN;
  __shared__ __align__(16) float As[128 * ASTRIDE];
  __shared__ __align__(16) float Bs[16 * 130 * 2];
  const int bi = (blockIdx.x >> 2) * 128;
  const int bj = (blockIdx.x & 3) * 128;
  const int tid = threadIdx.x;
  const int lane = tid & 31;
  const int w = tid >> 5;
  const int wr = (w & 3) * 32;
  const int wc = (w >> 2) * 64;
  v8f acc[2][4] = {};

  for (int k0 = 0; k0 < d; k0 += 32) {
    __syncthreads();
    fill_A_panel(As, A, bi, k0, d, tid);
    fill_B_panel<128, 130>(Bs, B, k0, bj, d, tid);
    if (k0 + 32 < d) {  // prefetch next panels -> global_prefetch_b8
      __builtin_prefetch(&A[(size_t)(bi + (tid & 127)) * d + k0 + 32], 0, 3);
      __builtin_prefetch(&B[(size_t)(k0 + 32 + (tid & 31)) * d + bj], 0, 3);
    }
    __syncthreads();
#pragma unroll
    for (int kk = 0; kk < 32; kk += 4)
      mma_step<2, 4, 130>(As, Bs, lane, kk, wr, wc, acc);
  }

#pragma unroll
  for (int rt = 0; rt < 2; ++rt)
#pragma unroll
    for (int ct = 0; ct < 4; ++ct)
#pragma unroll
      for (int r = 0; r < 8; ++r) {
        int row = bi + wr + rt * 16 + r + ((lane >> 4) << 3);
        int col = bj + wc + ct * 16 + (lane & 15);
        float v = alpha * acc[rt][ct][r];
        if (row == col) v += betaI;
        if (addMat) v += addScale * addMat[(size_t)row * d + col];
        D[(size_t)row * d + col] = v;
      }
}

// ---------------------------------------------------------------------------
// B_bar = dt * (P @ B_ct)   [512x32]   grid = 4 WGs (128 rows each)
// wave = 16 rows x 32 cols (1x2 tiles)
// ---------------------------------------------------------------------------
__global__ __launch_bounds__(256) void bbar_kernel(
    const float* __restrict__ P, const float* __restrict__ Bct,
    const float* __restrict__ dt, float* __restrict__ Bbar) {
  const int d = D_DYN;
  __shared__ __align__(16) float As[128 * ASTRIDE];
  __shared__ __align__(16) float Bs[16 * 34 * 2];
  const int bi = blockIdx.x * 128;
  const int tid = threadIdx.x;
  const int lane = tid & 31;
  const int w = tid >> 5;
  const int wr = w * 16;
  v8f acc[1][2] = {};
  const float dtv = dt[0];

  for (int k0 = 0; k0 < d; k0 += 32) {
    __syncthreads();
    fill_A_panel(As, P, bi, k0, d, tid);
    fill_B_panel<32, 34>(Bs, Bct, k0, 0, U_DIM, tid);
    __syncthreads();
#pragma unroll
    for (int kk = 0; kk < 32; kk += 4)
      mma_step<1, 2, 34>(As, Bs, lane, kk, wr, 0, acc);
  }
#pragma unroll
  for (int ct = 0; ct < 2; ++ct)
#pragma unroll
    for (int r = 0; r < 8; ++r) {
      int row = bi + wr + r + ((lane >> 4) << 3);
      int col = ct * 16 + (lane & 15);
      Bbar[(size_t)row * U_DIM + col] = dtv * acc[0][ct][r];
    }
}

// ---------------------------------------------------------------------------
// Scan step (state): zout = zprev @ AbarT + Ut @ BbarT   (both pre-transposed,
// row-major [K][512]).  grid = 16 WGs (128x128 tiles), wave = 32x64.
// ---------------------------------------------------------------------------
__global__ __launch_bounds__(256) void scan_z_kernel(
    const float* __restrict__ zprev, const float* __restrict__ AbarT,
    const float* __restrict__ Ut, const float* __restrict__ BbarT,
    float* __restrict__ zout) {
  const int d = D_DYN;
  __shared__ __align__(16) float As[128 * ASTRIDE];
  __shared__ __align__(16) float Bs[16 * 130 * 2];
  const int bi = (blockIdx.x >> 2) * 128;
  const int bj = (blockIdx.x & 3) * 128;
  const int tid = threadIdx.x;
  const int lane = tid & 31;
  const int w = tid >> 5;
  const int wr = (w & 3) * 32;
  const int wc = (w >> 2) * 64;
  v8f acc[2][4] = {};

  // phase 1: zprev @ AbarT  (K = 512)
  for (int k0 = 0; k0 < d; k0 += 32) {
    __syncthreads();
    fill_A_panel(As, zprev, bi, k0, d, tid);
    fill_B_panel<128, 130>(Bs, AbarT, k0, bj, d, tid);
    if (k0 + 32 < d) {
      __builtin_prefetch(&zprev[(size_t)(bi + (tid & 127)) * d + k0 + 32], 0, 3);
      __builtin_prefetch(&AbarT[(size_t)(k0 + 32 + (tid & 31)) * d + bj], 0, 3);
    }
    __syncthreads();
#pragma unroll
    for (int kk = 0; kk < 32; kk += 4)
      mma_step<2, 4, 130>(As, Bs, lane, kk, wr, wc, acc);
  }

  // phase 2: + Ut @ BbarT  (K = 32, single chunk)
  __syncthreads();
  fill_A_panel(As, Ut, bi, 0, U_DIM, tid);
  fill_B_panel<128, 130>(Bs, BbarT, 0, bj, d, tid);
  __syncthreads();
#pragma unroll
  for (int kk = 0; kk < 32; kk += 4)
    mma_step<2, 4, 130>(As, Bs, lane, kk, wr, wc, acc);

#pragma unroll
  for (int rt = 0; rt < 2; ++rt)
#pragma unroll
    for (int ct = 0; ct < 4; ++ct)
#pragma unroll
      for (int r = 0; r < 8; ++r) {
        int row = bi + wr + rt * 16 + r + ((lane >> 4) << 3);
        int col = bj + wc + ct * 16 + (lane & 15);
        zout[(size_t)row * d + col] = acc[rt][ct][r];
      }
}

// ---------------------------------------------------------------------------
// Scan step (obs): Y = Zt @ CT + Ut @ DTs   (CT: [512][64] zero-padded,
// DTs: [32][64] pre-scaled by dt).  grid = 4 WGs (128x64), wave = 32x32.
// ---------------------------------------------------------------------------
__global__ __launch_bounds__(256) void scan_y_kernel(
    const float* __restrict__ Zt, const float* __restrict__ Ut,
    const float* __restrict__ CT, const float* __restrict__ DTs,
    float* __restrict__ Y) {
  const int d = D_DYN;
  __shared__ __align__(16) float As[128 * ASTRIDE];
  __shared__ __align__(16) float Bs[16 * 66 * 2];
  const int bi = blockIdx.x * 128;
  const int tid = threadIdx.x;
  const int lane = tid & 31;
  const int w = tid >> 5;
  const int wr = (w & 3) * 32;
  const int wc = (w >> 2) * 32;
  v8f acc[2][2] = {};

  // phase 1: Zt @ CT  (K = 512)
  for (int k0 = 0; k0 < d; k0 += 32) {
    __syncthreads();
    fill_A_panel(As, Zt, bi, k0, d, tid);
    fill_B_panel<64, 66>(Bs, CT, k0, 0, NPAD, tid);
    if (k0 + 32 < d)
      __builtin_prefetch(&Zt[(size_t)(bi + (tid & 127)) * d + k0 + 32], 0, 3);
    __syncthreads();
#pragma unroll
    for (int kk = 0; kk < 32; kk += 4)
      mma_step<2, 2, 66>(As, Bs, lane, kk, wr, wc, acc);
  }

  // phase 2: + Ut @ (dt*D^T)  (K = 32, single chunk)
  __syncthreads();
  fill_A_panel(As, Ut, bi, 0, U_DIM, tid);
  fill_B_panel<64, 66>(Bs, DTs, 0, 0, NPAD, tid);
  __syncthreads();
#pragma unroll
  for (int kk = 0; kk < 32; kk += 4)
    mma_step<2, 2, 66>(As, Bs, lane, kk, wr, wc, acc);

#pragma unroll
  for (int rt = 0; rt < 2; ++rt)
#pragma unroll
    for (int ct = 0; ct < 2; ++ct)
#pragma unroll
      for (int r = 0; r < 8; ++r) {
        int row = bi + wr + rt * 16 + r + ((lane >> 4) << 3);
        int col = wc + ct * 16 + (lane & 15);
        if (col < N_OBS) Y[(size_t)row * N_OBS + col] = acc[rt][ct][r];
      }
}

// ---------------------------------------------------------------------------
// Host orchestration
// ---------------------------------------------------------------------------
extern "C" void kernel_launch(void* const* d_in, const int* in_sizes, int n_in,
                              void* d_out, int out_size, void* d_ws, size_t ws_size,
                              hipStream_t stream) {
  (void)in_sizes; (void)n_in; (void)out_size; (void)ws_size;
  const float* z_dyn = (const float*)d_in[0];  // [512,512]
  const float* dt    = (const float*)d_in[2];  // [1]
  const float* U     = (const float*)d_in[3];  // [256,512,32]
  const float* skew  = (const float*)d_in[4];  // [d(d-1)/2]
  const float* gamma = (const float*)d_in[5];  // [512]
  const float* Bct   = (const float*)d_in[6];  // [512,32]
  const float* Cm    = (const float*)d_in[7];  // [50,512]
  const float* Dm    = (const float*)d_in[8];  // [50,32]

  const size_t DD = (size_t)D_DYN * D_DYN;
  float* Mb    = (float*)d_ws;
  float* E[2]  = {Mb + DD, Mb + 2 * DD};
  float* P[2]  = {Mb + 3 * DD, Mb + 4 * DD};
  float* AbarT = Mb + 5 * DD;                       // [512][512]
  float* Bbar  = Mb + 6 * DD;                       // [512][32]
  float* BbarT = Bbar + (size_t)D_DYN * U_DIM;      // [32][512]
  float* CT    = BbarT + (size_t)U_DIM * D_DYN;     // [512][64]
  float* DTs   = CT + (size_t)D_DYN * NPAD;         // [32][64]

  float* Zout = (float*)d_out;                           // [256,512,512]
  float* Yout = Zout + (size_t)T_STEPS * BATCH * D_DYN;  // [256,512,50]

  const dim3 blk(256);
  const int grid_ew = (int)(DD / 256);

  // Phase 1: M = A_ct * dt / 2^NSQ
  build_M_kernel<<<grid_ew, blk, 0, stream>>>(skew, gamma, dt, Mb);

  // Phase 2a: E = exp(M), order-8 Taylor via Horner (ping-pong E0/E1)
  set_identity_kernel<<<grid_ew, blk, 0, stream>>>(E[0], 1.0f);
  int ec = 0;
  for (int k = 8; k >= 1; --k) {
    gemm512_kernel<<<16, blk, 0, stream>>>(Mb, E[ec], E[1 - ec],
                                           1.0f / (float)k, 1.0f, nullptr, 0.0f);
    ec = 1 - ec;
  }
  // Phase 2b: P = phi1(M) = sum_k M^k/(k+1)!, order 8
  const float ck[9] = {1.0f, 1.0f / 2, 1.0f / 6, 1.0f / 24, 1.0f / 120,
                       1.0f / 720, 1.0f / 5040, 1.0f / 40320, 1.0f / 362880};
  set_identity_kernel<<<grid_ew, blk, 0, stream>>>(P[0], ck[8]);
  int pc = 0;
  for (int k = 7; k >= 0; --k) {
    gemm512_kernel<<<16, blk, 0, stream>>>(Mb, P[pc], P[1 - pc],
                                           1.0f, ck[k], nullptr, 0.0f);
    pc = 1 - pc;
  }
  // Phase 2c: doubling — P <- P(E+I)/2 (uses old E), then E <- E^2
  for (int i = 0; i < NSQ; ++i) {
    gemm512_kernel<<<16, blk, 0, stream>>>(P[pc], E[ec], P[1 - pc],
                                           0.5f, 0.0f, P[pc], 0.5f);
    gemm512_kernel<<<16, blk, 0, stream>>>(E[ec], E[ec], E[1 - ec],
                                           1.0f, 0.0f, nullptr, 0.0f);
    pc = 1 - pc; ec = 1 - ec;
  }
  // B_bar = dt * P @ B_ct
  bbar_kernel<<<4, blk, 0, stream>>>(P[pc], Bct, dt, Bbar);

  // One-time operator transposes / padding for the scan
  transpose_pad_kernel<<<1024, blk, 0, stream>>>(E[ec], AbarT, D_DYN, D_DYN,
                                                 D_DYN, nullptr);
  transpose_pad_kernel<<<64, blk, 0, stream>>>(Bbar, BbarT, D_DYN, U_DIM,
                                               D_DYN, nullptr);
  transpose_pad_kernel<<<128, blk, 0, stream>>>(Cm, CT, N_OBS, D_DYN,
                                                NPAD, nullptr);
  transpose_pad_kernel<<<8, blk, 0, stream>>>(Dm, DTs, N_OBS, U_DIM,
                                              NPAD, dt);

  // Phase 3: sequential scan, one Z kernel + one Y kernel per step
  for (int t = 0; t < T_STEPS; ++t) {
    const float* zprev =
        (t == 0) ? z_dyn : (Zout + (size_t)(t - 1) * BATCH * D_DYN);
    const float* Ut = U + (size_t)t * BATCH * U_DIM;
    float* zt = Zout + (size_t)t * BATCH * D_DYN;
    scan_z_kernel<<<16, blk, 0, stream>>>(zprev, AbarT, Ut, BbarT, zt);
    scan_y_kernel<<<4, blk, 0, stream>>>(zt, Ut, CT, DTs,
                                         Yout + (size_t)t * BATCH * N_OBS);
  }
}